// DeformConv2d_34505767256661
// MI455X (gfx1250) — compile-verified
//
#include <hip/hip_runtime.h>
#include <cstdint>

// DCNv2 modulated deformable sampling, NHWC, f32.
// B=1, H=100, W=160, G=8, C=64, K=9 (3x3, pad=1, stride=1, dil=1).
#define H_ 100
#define W_ 160
#define G_ 8
#define C_ 64
#define K_ 9

static __device__ __forceinline__ float rl_f(float v, int l) {
    return __uint_as_float(__builtin_amdgcn_readlane(__float_as_uint(v), l));
}
static __device__ __forceinline__ int rl_i(int v, int l) {
    return __builtin_amdgcn_readlane(v, l);
}

__global__ __launch_bounds__(256, 4)
void DeformConv2d_34505767256661_kernel(const float* __restrict__ x,
                                        const float* __restrict__ offset,
                                        const float* __restrict__ mask,
                                        float* __restrict__ out) {
    // One block = one output pixel. 8 waves (wave32) = 8 groups.
    __shared__ float smem[216];   // [0..143] offsets, [144..215] mask logits

    const int tid = threadIdx.x;
    const int pix = blockIdx.x;            // h*W + w
    const int h   = pix / W_;
    const int w   = pix - h * W_;

    // ---------------- async-stage offset(144)+mask(72) for this pixel into LDS
    // One dword per lane via CDNA5 async global->LDS DMA (ASYNCcnt).
    // LDS dest address is derived from &smem[tid]: for flat pointers in the
    // LDS aperture, addr[31:0] IS the LDS byte offset (ISA 10.2). Passing the
    // ptrtoint'd address into the asm also captures `smem`, so the "memory"
    // clobber keeps the later smem reads from being folded to poison.
    if (tid < 216) {
        const float* src = (tid < 144)
            ? (offset + (size_t)pix * (G_ * K_ * 2) + tid)
            : (mask   + (size_t)pix * (G_ * K_)     + (tid - 144));
        unsigned lds_byte = (unsigned)(uintptr_t)(&smem[tid]);
        asm volatile("global_load_async_to_lds_b32 %0, %1, off"
                     :: "v"(lds_byte), "v"(src)
                     : "memory");
    }
    asm volatile("s_wait_asynccnt 0x0" ::: "memory");
    __syncthreads();

    const int g    = tid >> 5;    // wave id == group id
    const int lane = tid & 31;

    // ---------------- phase A: lanes 0..8 each own one tap k ------------------
    // (lanes 9..31 compute tap 0 redundantly; their results are never read)
    const int   kk  = (lane < K_) ? lane : 0;
    const float mk  = smem[144 + g * K_ + kk];
    const float ofh = smem[g * (K_ * 2) + 2 * kk + 0];
    const float ofw = smem[g * (K_ * 2) + 2 * kk + 1];
    const float e   = __expf(mk);             // softmax numerator (denom folded at end)

    const int kyi = kk / 3;
    const int kxi = kk - kyi * 3;
    const float loc_h = (float)h + (float)(kyi - 1) + ofh;
    const float loc_w = (float)w + (float)(kxi - 1) + ofw;
    const float fh0 = floorf(loc_h);
    const float fw0 = floorf(loc_w);
    const float th  = loc_h - fh0;
    const float tw  = loc_w - fw0;
    const int h0 = (int)fh0, w0 = (int)fw0;
    const int h1 = h0 + 1,   w1 = w0 + 1;

    const bool vh0 = (h0 >= 0) & (h0 < H_);
    const bool vh1 = (h1 >= 0) & (h1 < H_);
    const bool vw0 = (w0 >= 0) & (w0 < W_);
    const bool vw1 = (w1 >= 0) & (w1 < W_);
    const int ch0 = min(max(h0, 0), H_ - 1);
    const int ch1 = min(max(h1, 0), H_ - 1);
    const int cw0 = min(max(w0, 0), W_ - 1);
    const int cw1 = min(max(w1, 0), W_ - 1);

    // fold exp(mask) and validity into the bilinear corner weights
    const float w00 = (1.f - th) * (1.f - tw) * ((vh0 && vw0) ? e : 0.f);
    const float w01 = (1.f - th) * tw         * ((vh0 && vw1) ? e : 0.f);
    const float w10 = th         * (1.f - tw) * ((vh1 && vw0) ? e : 0.f);
    const float w11 = th         * tw         * ((vh1 && vw1) ? e : 0.f);

    // corner base offsets in float2 units (channel-pair granularity)
    const int gc  = g * C_;
    const int o00 = (((ch0 * W_ + cw0) * (G_ * C_)) + gc) >> 1;
    const int o01 = (((ch0 * W_ + cw1) * (G_ * C_)) + gc) >> 1;
    const int o10 = (((ch1 * W_ + cw0) * (G_ * C_)) + gc) >> 1;
    const int o11 = (((ch1 * W_ + cw1) * (G_ * C_)) + gc) >> 1;

    // ---------------- phase B: broadcast tap scalars, gather channels ---------
    // lane owns channels (2*lane, 2*lane+1) -> coalesced global_load_b64.
    const float2* __restrict__ x2 = (const float2*)x;
    float ax = 0.f, ay = 0.f, esum = 0.f;

    #pragma unroll
    for (int k = 0; k < K_; ++k) {
        esum += rl_f(e, k);
        const float b00 = rl_f(w00, k);
        const float b01 = rl_f(w01, k);
        const float b10 = rl_f(w10, k);
        const float b11 = rl_f(w11, k);
        const int   i00 = rl_i(o00, k) + lane;   // SGPR base + lane offset
        const int   i01 = rl_i(o01, k) + lane;
        const int   i10 = rl_i(o10, k) + lane;
        const int   i11 = rl_i(o11, k) + lane;

        const float2 a0 = x2[i00];
        const float2 a1 = x2[i01];
        const float2 a2 = x2[i10];
        const float2 a3 = x2[i11];

        ax = fmaf(b00, a0.x, ax);  ay = fmaf(b00, a0.y, ay);
        ax = fmaf(b01, a1.x, ax);  ay = fmaf(b01, a1.y, ay);
        ax = fmaf(b10, a2.x, ax);  ay = fmaf(b10, a2.y, ay);
        ax = fmaf(b11, a3.x, ax);  ay = fmaf(b11, a3.y, ay);
    }

    const float rs = 1.0f / esum;               // softmax normalization
    float2* __restrict__ out2 = (float2*)out;
    const int oi = pix * (G_ * C_ / 2) + g * (C_ / 2) + lane;
    out2[oi] = make_float2(ax * rs, ay * rs);
}

extern "C" void kernel_launch(void* const* d_in, const int* in_sizes, int n_in,
                              void* d_out, int out_size, void* d_ws, size_t ws_size,
                              hipStream_t stream) {
    (void)in_sizes; (void)n_in; (void)out_size; (void)d_ws; (void)ws_size;
    const float* x      = (const float*)d_in[0];
    const float* offset = (const float*)d_in[1];
    const float* mask   = (const float*)d_in[2];
    float* out          = (float*)d_out;

    dim3 grid(H_ * W_);     // one block per pixel
    dim3 block(256);        // 8 wave32 = 8 groups
    DeformConv2d_34505767256661_kernel<<<grid, block, 0, stream>>>(x, offset, mask, out);
}